// FNO_RC_2D_56358560858537
// MI455X (gfx1250) — compile-verified
//
#include <hip/hip_runtime.h>
#include <math.h>
#include <stdint.h>

// ---------------- problem constants ----------------
#define BATCH 8
#define CH    64
#define HP    128            // padded H == W
#define NPIX  (HP*HP)        // 16384
#define SDIM  119
#define KY    16             // kept ky modes
#define KX    32             // kept kx modes (0..15, 112..127)
#define NOUT  (BATCH*SDIM*SDIM)
#define PI_F  3.14159265358979323846f

typedef __attribute__((ext_vector_type(2))) float v2f;
typedef __attribute__((ext_vector_type(8))) float v8f;

__device__ __forceinline__ v8f wmma4(v2f a, v2f b, v8f c) {
  // D = A(16x4,f32) * B(4x16,f32) + C(16x16,f32)
  return __builtin_amdgcn_wmma_f32_16x16x4_f32(false, a, false, b, (short)0, c,
                                               false, false);
}
__device__ __forceinline__ float gelu_exact(float x) {
  return 0.5f * x * (1.0f + erff(x * 0.70710678118654752f));
}

// -------- CDNA5 async global->LDS copy (ASYNCcnt tracked) --------
// LDS destination offset = low 32 bits of the generic shared-memory address
__device__ __forceinline__ void async_copy_b128(void* lds_dst, const void* g_src) {
  unsigned l = (unsigned)(uintptr_t)lds_dst;
  asm volatile("global_load_async_to_lds_b128 %0, %1, off"
               :: "v"(l), "v"(g_src) : "memory");
}
__device__ __forceinline__ void async_copy_b32(void* lds_dst, const void* g_src) {
  unsigned l = (unsigned)(uintptr_t)lds_dst;
  asm volatile("global_load_async_to_lds_b32 %0, %1, off"
               :: "v"(l), "v"(g_src) : "memory");
}
__device__ __forceinline__ void async_wait0() {
  asm volatile("s_wait_asynccnt 0x0" ::: "memory");
}

// ---------------- twiddle/matrix construction ----------------
// Wy [128][32]  : forward y-DFT, cols 0..15 = cos, 16..31 = -sin
// WxA[32][256]  : forward x-DFT realified A' rows kx-idx, pairs (cos, -sin)
// ExA[128][64]  : inverse x-DFT realified A' rows x, pairs (cos, +sin)
// Ey [32][128]  : inverse y rfft as B matrix, rows 0..15 = w*cos, 16..31 = -w*sin
//                 (w = 1 for k=0 else 2, scaled by 1/(H*W))
__global__ void k_twiddles(float* Wy, float* WxA, float* ExA, float* Ey) {
  int tid = blockIdx.x * blockDim.x + threadIdx.x;
  int nthr = gridDim.x * blockDim.x;
  const float step = 2.0f * PI_F / (float)HP;
  for (int i = tid; i < HP * 32; i += nthr) {
    int w = i >> 5, c = i & 31;
    int k = c & 15;
    float th = step * (float)((w * k) & (HP - 1));
    Wy[i] = (c < 16) ? cosf(th) : -sinf(th);
  }
  for (int i = tid; i < 32 * 256; i += nthr) {
    int m = i >> 8, r = i & 255;
    int x = r >> 1, p = r & 1;
    int kx = (m < 16) ? m : (m + 96);
    float th = step * (float)((kx * x) & (HP - 1));
    WxA[i] = p ? -sinf(th) : cosf(th);
  }
  for (int i = tid; i < HP * 64; i += nthr) {
    int x = i >> 6, r = i & 63;
    int m = r >> 1, p = r & 1;
    int kx = (m < 16) ? m : (m + 96);
    float th = step * (float)((kx * x) & (HP - 1));
    ExA[i] = p ? sinf(th) : cosf(th);
  }
  const float sc = 1.0f / ((float)HP * (float)HP);
  for (int i = tid; i < 32 * HP; i += nthr) {
    int c = i >> 7, y = i & 127;
    int k = (c < 16) ? c : (c - 16);
    float th = step * (float)((k * y) & (HP - 1));
    float wgt = ((k == 0) ? 1.0f : 2.0f) * sc;
    Ey[i] = (c < 16) ? wgt * cosf(th) : -wgt * sinf(th);
  }
}

// ---------------- lifting: fc0 + grid concat + pad ----------------
__global__ void k_fc0(const float* __restrict__ xin, const float* __restrict__ fc0w,
                      const float* __restrict__ fc0b, float* __restrict__ h) {
  int t = blockIdx.x * blockDim.x + threadIdx.x;
  if (t >= BATCH * NPIX) return;
  int b = t >> 14;
  int pix = t & (NPIX - 1);
  int y = pix >> 7, x = pix & 127;
  float in0 = 0.f, in1 = 0.f, in2 = 0.f, gx = 0.f, gy = 0.f;
  bool valid = (y < SDIM) && (x < SDIM);
  if (valid) {
    const float* xp = xin + ((size_t)(b * SDIM + y) * SDIM + x) * 3;
    in0 = xp[0]; in1 = xp[1]; in2 = xp[2];
    gx = (float)y / (float)(SDIM - 1);
    gy = (float)x / (float)(SDIM - 1);
  }
  float* hp = h + (size_t)b * CH * NPIX + pix;
  #pragma unroll 4
  for (int c = 0; c < CH; ++c) {
    float v = 0.f;
    if (valid) {
      const float* wr = fc0w + c * 5;
      v = wr[0]*in0 + wr[1]*in1 + wr[2]*in2 + wr[3]*gx + wr[4]*gy + fc0b[c];
    }
    hp[(size_t)c * NPIX] = v;
  }
}

// ---------------- forward y-DFT: yf[bcx][32] = h[bcx][128] @ Wy ----------------
__global__ void k_dft_y(const float* __restrict__ h, const float* __restrict__ Wy,
                        float* __restrict__ yf) {
  __shared__ float sWy[HP * 32];
  for (int i = threadIdx.x; i < (HP * 32) / 4; i += blockDim.x)
    async_copy_b128(&sWy[i * 4], Wy + i * 4);
  async_wait0();
  __syncthreads();

  int lane = threadIdx.x & 31, wave = threadIdx.x >> 5;
  int mtile = blockIdx.x * 4 + wave;          // 4096 tiles over 65536 rows
  int lh = lane >> 4, ln = lane & 15;
  const float* arow = h + (size_t)(mtile * 16 + ln) * HP;
  v8f acc0 = {}; v8f acc1 = {};
  for (int kt = 0; kt < 32; ++kt) {
    int kk = kt * 4 + lh * 2;
    v2f a;  a.x = arow[kk];               a.y = arow[kk + 1];
    v2f b0; b0.x = sWy[kk*32 + ln];       b0.y = sWy[(kk+1)*32 + ln];
    v2f b1; b1.x = sWy[kk*32 + 16 + ln];  b1.y = sWy[(kk+1)*32 + 16 + ln];
    acc0 = wmma4(a, b0, acc0);
    acc1 = wmma4(a, b1, acc1);
  }
  float* orow = yf + (size_t)(mtile * 16) * 32;
  #pragma unroll
  for (int r = 0; r < 8; ++r) {
    int m = r + lh * 8;
    orow[m*32 + ln]      = acc0[r];
    orow[m*32 + 16 + ln] = acc1[r];
  }
}

// ---------------- forward x-DFT (complex, realified K=256) ----------------
__global__ void k_dft_x(const float* __restrict__ yf, const float* __restrict__ WxA,
                        float* __restrict__ xf) {
  __shared__ float sWx[32 * 256];
  for (int i = threadIdx.x; i < (32 * 256) / 4; i += blockDim.x)
    async_copy_b128(&sWx[i * 4], WxA + i * 4);
  async_wait0();
  __syncthreads();

  int bc = blockIdx.x;                         // 512 (b*64+c)
  int lane = threadIdx.x & 31, wave = threadIdx.x >> 5;  // 4 waves
  int mtile = wave & 1, ntile = wave >> 1;
  int lh = lane >> 4, ln = lane & 15;
  int m = mtile * 16 + ln;
  const float* yb = yf + (size_t)bc * HP * 32;
  v8f acc = {};
  for (int kt = 0; kt < 64; ++kt) {
    int kk = kt * 4 + lh * 2;                  // even
    v2f a; a.x = sWx[m*256 + kk]; a.y = sWx[m*256 + kk + 1];
    int x = kk >> 1;
    float yr = yb[x*32 + ln];
    float yi = yb[x*32 + 16 + ln];
    v2f b;
    if (ntile == 0) { b.x = yr; b.y = -yi; }   // Re output columns
    else            { b.x = yi; b.y =  yr; }   // Im output columns
    acc = wmma4(a, b, acc);
  }
  float* ob = xf + (size_t)bc * KX * 32;
  #pragma unroll
  for (int r = 0; r < 8; ++r) {
    int mm = mtile * 16 + r + lh * 8;
    ob[mm*32 + ntile*16 + ln] = acc[r];
  }
}

// ---------------- per-mode channel mixing (complex GEMM 8x64x64) ----------------
__global__ void k_modemix(const float* __restrict__ xf,
                          const float* __restrict__ w1r, const float* __restrict__ w1i,
                          const float* __restrict__ w2r, const float* __restrict__ w2i,
                          float* __restrict__ mf) {
  __shared__ float sX[BATCH * CH * 2];         // (Xr, Xi) pairs for this mode
  int mode = blockIdx.x;                       // 512 modes
  int kxi = mode >> 4, ky = mode & 15;

  // per-lane async gather of the whole activation slice for this mode
  for (int idx = threadIdx.x; idx < BATCH * CH; idx += 32) {
    const float* xp = xf + ((size_t)idx * KX + kxi) * 32;
    async_copy_b32(&sX[idx * 2],     xp + ky);
    async_copy_b32(&sX[idx * 2 + 1], xp + 16 + ky);
  }
  async_wait0();
  __syncthreads();

  int lane = threadIdx.x & 31;
  int lh = lane >> 4, ln = lane & 15;
  const float* wr; const float* wi; int moff;
  if (kxi < 16) { wr = w1r; wi = w1i; moff = kxi*16 + ky; }
  else          { wr = w2r; wi = w2i; moff = (kxi - 16)*16 + ky; }
  v8f acc[8] = {};
  bool avalid = (ln < 8);
  for (int kt = 0; kt < 32; ++kt) {
    int kk = kt * 4 + lh * 2;                  // even; i = kk/2, pair (Xr, Xi)
    int i = kk >> 1;
    v2f a = {0.f, 0.f};
    if (avalid) {
      a.x = sX[(ln * CH + i) * 2];
      a.y = sX[(ln * CH + i) * 2 + 1];
    }
    #pragma unroll
    for (int nt = 0; nt < 8; ++nt) {
      int o = (nt & 3) * 16 + ln;
      float vr = wr[((size_t)(i * CH + o)) * 256 + moff];
      float vi = wi[((size_t)(i * CH + o)) * 256 + moff];
      v2f b;
      if (nt < 4) { b.x = vr; b.y = -vi; }     // Cr
      else        { b.x = vi; b.y =  vr; }     // Ci
      acc[nt] = wmma4(a, b, acc[nt]);
    }
  }
  if (lh == 0) {                               // rows 0..7 == batch 0..7
    #pragma unroll
    for (int nt = 0; nt < 8; ++nt) {
      int o = (nt & 3) * 16 + ln;
      #pragma unroll
      for (int r = 0; r < 8; ++r) {
        float* mp = mf + ((size_t)(r * CH + o) * KX + kxi) * 32;
        if (nt < 4) mp[ky]      = acc[nt][r];
        else        mp[16 + ky] = acc[nt][r];
      }
    }
  }
}

// ---------------- inverse x-DFT (complex, realified K=64) ----------------
__global__ void k_idft_x(const float* __restrict__ mf, const float* __restrict__ ExA,
                         float* __restrict__ zf) {
  __shared__ float sEx[HP * 64];
  __shared__ float sM[KX * 32];
  int bo = blockIdx.x;                         // 512 (b*64+o)
  const float* mb = mf + (size_t)bo * KX * 32;
  for (int i = threadIdx.x; i < (HP * 64) / 4; i += blockDim.x)
    async_copy_b128(&sEx[i * 4], ExA + i * 4);
  for (int i = threadIdx.x; i < (KX * 32) / 4; i += blockDim.x)
    async_copy_b128(&sM[i * 4], mb + i * 4);
  async_wait0();
  __syncthreads();

  int lane = threadIdx.x & 31, wave = threadIdx.x >> 5;  // 8 waves
  int lh = lane >> 4, ln = lane & 15;
  int x = wave * 16 + ln;
  v8f acc0 = {}; v8f acc1 = {};
  for (int kt = 0; kt < 16; ++kt) {
    int kk = kt * 4 + lh * 2;
    v2f a; a.x = sEx[x*64 + kk]; a.y = sEx[x*64 + kk + 1];
    int m = kk >> 1;
    float mr = sM[m*32 + ln];
    float mi = sM[m*32 + 16 + ln];
    v2f b0; b0.x = mr; b0.y = -mi;
    v2f b1; b1.x = mi; b1.y =  mr;
    acc0 = wmma4(a, b0, acc0);
    acc1 = wmma4(a, b1, acc1);
  }
  float* ob = zf + (size_t)bo * HP * 32;
  #pragma unroll
  for (int r = 0; r < 8; ++r) {
    int xx = wave * 16 + r + lh * 8;
    ob[xx*32 + ln]      = acc0[r];
    ob[xx*32 + 16 + ln] = acc1[r];
  }
}

// ---------------- channel mix GEMM: h2 = cw @ h + cb ----------------
__global__ void k_chanmix(const float* __restrict__ h, const float* __restrict__ cw,
                          const float* __restrict__ cb, float* __restrict__ h2) {
  __shared__ float scw[CH * CH];
  for (int i = threadIdx.x; i < (CH * CH) / 4; i += blockDim.x)
    async_copy_b128(&scw[i * 4], cw + i * 4);
  async_wait0();
  __syncthreads();

  int lane = threadIdx.x & 31, wave = threadIdx.x >> 5;  // 8 waves
  int mtile = blockIdx.x * 8 + wave;            // 8192 tiles over 131072 pixel-rows
  int lh = lane >> 4, ln = lane & 15;
  int g = mtile * 16;
  int b = g >> 14;
  int pix = g & (NPIX - 1);
  const float* hb = h + (size_t)b * CH * NPIX + pix + ln;
  v8f acc[4] = {};
  for (int kt = 0; kt < 16; ++kt) {
    int kk = kt * 4 + lh * 2;
    if (kt + 2 < 16)
      __builtin_prefetch(hb + (size_t)(kt * 4 + 8) * NPIX, 0, 0);
    v2f a; a.x = hb[(size_t)kk * NPIX]; a.y = hb[(size_t)(kk + 1) * NPIX];
    #pragma unroll
    for (int nt = 0; nt < 4; ++nt) {
      int o = nt * 16 + ln;
      v2f bf; bf.x = scw[o*CH + kk]; bf.y = scw[o*CH + kk + 1];
      acc[nt] = wmma4(a, bf, acc[nt]);
    }
  }
  #pragma unroll
  for (int nt = 0; nt < 4; ++nt) {
    int o = nt * 16 + ln;
    float bias = cb[o];
    float* op = h2 + (size_t)(b * CH + o) * NPIX + pix;
    #pragma unroll
    for (int r = 0; r < 8; ++r) op[r + lh * 8] = acc[nt][r] + bias;
  }
}

// ---------------- inverse y + residual (h2) + corr + optional GELU ----------------
__global__ void k_combine(const float* __restrict__ zf, const float* __restrict__ Ey,
                          const float* __restrict__ h2, const float* __restrict__ corr,
                          float* __restrict__ hout, int apply_gelu) {
  __shared__ float sEy[32 * HP];
  for (int i = threadIdx.x; i < (32 * HP) / 4; i += blockDim.x)
    async_copy_b128(&sEy[i * 4], Ey + i * 4);
  async_wait0();
  __syncthreads();

  int lane = threadIdx.x & 31, wave = threadIdx.x >> 5;  // 4 waves
  int mtile = blockIdx.x * 4 + wave;            // 4096 tiles over 65536 rows
  int lh = lane >> 4, ln = lane & 15;
  int row0 = mtile * 16;
  const float* arow = zf + (size_t)(row0 + ln) * 32;
  v8f acc[8] = {};
  for (int kt = 0; kt < 8; ++kt) {
    int kk = kt * 4 + lh * 2;
    v2f a; a.x = arow[kk]; a.y = arow[kk + 1];
    #pragma unroll
    for (int nt = 0; nt < 8; ++nt) {
      int y = nt * 16 + ln;
      v2f b; b.x = sEy[kk*HP + y]; b.y = sEy[(kk+1)*HP + y];
      acc[nt] = wmma4(a, b, acc[nt]);
    }
  }
  #pragma unroll
  for (int r = 0; r < 8; ++r) {
    int g = row0 + r + lh * 8;                  // (b*64+o)*128 + x
    int bo = g >> 7, x = g & 127;
    float cr = corr[bo];
    const float* h2p = h2 + (size_t)bo * NPIX + x * HP;
    float* hop = hout + (size_t)bo * NPIX + x * HP;
    #pragma unroll
    for (int nt = 0; nt < 8; ++nt) {
      int y = nt * 16 + ln;
      float v = acc[nt][r] + h2p[y] + cr;
      if (apply_gelu) v = gelu_exact(v);
      hop[y] = v;
    }
  }
}

// ---------------- CFT coefficients (only segments (0,0),(0,1) survive) ----------------
__global__ void k_cft(const float* __restrict__ h, float* __restrict__ cflat) {
  __shared__ float sred[256];
  __shared__ float ssq[256];
  int bc = blockIdx.x;
  int b = bc >> 6, c = bc & 63;
  int t = threadIdx.x;
  const float* hb = h + (size_t)bc * NPIX;
  for (int seg = 0; seg < 2; ++seg) {
    float ss = 0.f, sm = 0.f;
    for (int e = t; e < 1024; e += 256) {       // e = hi*32+wi; e mod 8 == t mod 8
      int hi = e >> 5, wi = e & 31;
      float v = hb[hi * HP + seg * 32 + wi];
      ss += v * v;
      sm += v;
    }
    ssq[t] = ss; sred[t] = sm;
    __syncthreads();
    for (int st = 128; st >= 8; st >>= 1) {     // preserves mod-8 classes
      if (t < st) { sred[t] += sred[t + st]; ssq[t] += ssq[t + st]; }
      __syncthreads();
    }
    if (t < 8) {
      float tot = ssq[0]+ssq[1]+ssq[2]+ssq[3]+ssq[4]+ssq[5]+ssq[6]+ssq[7];
      float nrm = fmaxf(sqrtf(tot), 1e-12f);
      float coeff = (sred[t] * (1.0f / 128.0f)) / nrm;
      int idx = c * 16 + seg * 8 + t;
      cflat[(size_t)b * 2048 + idx * 2]     = coeff;  // real
      cflat[(size_t)b * 2048 + idx * 2 + 1] = 0.f;    // imag
    }
    __syncthreads();
  }
}

// ---------------- g1: [8,2048] -> [8,256], GELU ----------------
__global__ void k_g1(const float* __restrict__ cflat, const float* __restrict__ g1w,
                     const float* __restrict__ g1b, float* __restrict__ h1g) {
  int nt = blockIdx.x;                          // 16 tiles of N=256
  int lane = threadIdx.x & 31;
  int lh = lane >> 4, ln = lane & 15;
  int n = nt * 16 + ln;
  const float* brow = g1w + (size_t)n * 2048;
  v8f acc = {};
  for (int kt = 0; kt < 512; ++kt) {
    int kk = kt * 4 + lh * 2;
    v2f a = {0.f, 0.f};
    if (ln < 8) {
      const float* arow = cflat + (size_t)ln * 2048;
      a.x = arow[kk]; a.y = arow[kk + 1];
    }
    v2f bf; bf.x = brow[kk]; bf.y = brow[kk + 1];
    acc = wmma4(a, bf, acc);
  }
  if (lh == 0) {
    float bias = g1b[n];
    #pragma unroll
    for (int r = 0; r < 8; ++r) h1g[r * 256 + n] = gelu_exact(acc[r] + bias);
  }
}

// ---------------- g2: [8,256] -> [8,64] corr ----------------
__global__ void k_g2(const float* __restrict__ h1g, const float* __restrict__ g2w,
                     const float* __restrict__ g2b, float* __restrict__ corr) {
  int lane = threadIdx.x & 31, wave = threadIdx.x >> 5;  // 4 waves
  int lh = lane >> 4, ln = lane & 15;
  int n = wave * 16 + ln;
  v8f acc = {};
  for (int kt = 0; kt < 64; ++kt) {
    int kk = kt * 4 + lh * 2;
    v2f a = {0.f, 0.f};
    if (ln < 8) { a.x = h1g[ln * 256 + kk]; a.y = h1g[ln * 256 + kk + 1]; }
    v2f bf; bf.x = g2w[n * 256 + kk]; bf.y = g2w[n * 256 + kk + 1];
    acc = wmma4(a, bf, acc);
  }
  if (lh == 0) {
    float bias = g2b[n];
    #pragma unroll
    for (int r = 0; r < 8; ++r) corr[r * CH + n] = acc[r] + bias;
  }
}

// ---------------- head: crop + fc1 (WMMA) + GELU + fc2 (shuffle dot) ----------------
__global__ void k_head(const float* __restrict__ h, const float* __restrict__ fc1w,
                       const float* __restrict__ fc1b, const float* __restrict__ fc2w,
                       const float* __restrict__ fc2b, float* __restrict__ out) {
  __shared__ float sW1[128 * CH];
  for (int i = threadIdx.x; i < (128 * CH) / 4; i += blockDim.x)
    async_copy_b128(&sW1[i * 4], fc1w + i * 4);
  async_wait0();
  __syncthreads();

  int lane = threadIdx.x & 31, wave = threadIdx.x >> 5;  // 4 waves
  int mtile = blockIdx.x * 4 + wave;
  if (mtile * 16 >= NOUT) return;               // wave-uniform exit
  int lh = lane >> 4, ln = lane & 15;
  int rowA = mtile * 16 + ln;
  bool avalid = rowA < NOUT;
  int b = 0, y = 0, x = 0;
  if (avalid) {
    b = rowA / (SDIM * SDIM);
    int rem = rowA - b * (SDIM * SDIM);
    y = rem / SDIM; x = rem - y * SDIM;
  }
  const float* hp = h + (size_t)b * CH * NPIX + y * HP + x;
  v8f acc[8] = {};
  for (int kt = 0; kt < 16; ++kt) {
    int kk = kt * 4 + lh * 2;
    v2f a = {0.f, 0.f};
    if (avalid) { a.x = hp[(size_t)kk * NPIX]; a.y = hp[(size_t)(kk + 1) * NPIX]; }
    #pragma unroll
    for (int nt = 0; nt < 8; ++nt) {
      int j = nt * 16 + ln;
      v2f bf; bf.x = sW1[j * CH + kk]; bf.y = sW1[j * CH + kk + 1];
      acc[nt] = wmma4(a, bf, acc[nt]);
    }
  }
  float f2b = fc2b[0];
  #pragma unroll
  for (int r = 0; r < 8; ++r) {
    float s = 0.f;
    #pragma unroll
    for (int nt = 0; nt < 8; ++nt) {
      int j = nt * 16 + ln;
      s += gelu_exact(acc[nt][r] + fc1b[j]) * fc2w[j];
    }
    // reduce the 16 column-lanes (xor masks < 16 stay within each half)
    s += __shfl_xor(s, 1);
    s += __shfl_xor(s, 2);
    s += __shfl_xor(s, 4);
    s += __shfl_xor(s, 8);
    int row = mtile * 16 + r + lh * 8;
    if (ln == 0 && row < NOUT) out[row] = s + f2b;
  }
}

// ---------------- host launcher ----------------
extern "C" void kernel_launch(void* const* d_in, const int* in_sizes, int n_in,
                              void* d_out, int out_size, void* d_ws, size_t ws_size,
                              hipStream_t stream) {
  const float* xin  = (const float*)d_in[0];
  const float* sw1r = (const float*)d_in[1];
  const float* sw1i = (const float*)d_in[2];
  const float* sw2r = (const float*)d_in[3];
  const float* sw2i = (const float*)d_in[4];
  const float* g1w  = (const float*)d_in[5];
  const float* g1b  = (const float*)d_in[6];
  const float* g2w  = (const float*)d_in[7];
  const float* g2b  = (const float*)d_in[8];
  const float* cw   = (const float*)d_in[9];
  const float* cb   = (const float*)d_in[10];
  const float* fc0w = (const float*)d_in[11];
  const float* fc0b = (const float*)d_in[12];
  const float* fc1w = (const float*)d_in[13];
  const float* fc1b = (const float*)d_in[14];
  const float* fc2w = (const float*)d_in[15];
  const float* fc2b = (const float*)d_in[16];

  float* ws = (float*)d_ws;
  size_t o = 0;
  float* Wy   = ws + o; o += (size_t)HP * 32;
  float* WxA  = ws + o; o += (size_t)32 * 256;
  float* ExA  = ws + o; o += (size_t)HP * 64;
  float* Ey   = ws + o; o += (size_t)32 * HP;
  float* hA   = ws + o; o += (size_t)BATCH * CH * NPIX;
  float* hB   = ws + o; o += (size_t)BATCH * CH * NPIX;
  float* yfb  = ws + o; o += (size_t)BATCH * CH * HP * 32;
  float* xfb  = ws + o; o += (size_t)BATCH * CH * KX * 32;
  float* mfb  = ws + o; o += (size_t)BATCH * CH * KX * 32;
  float* zfb  = ws + o; o += (size_t)BATCH * CH * HP * 32;
  float* h2b  = ws + o; o += (size_t)BATCH * CH * NPIX;
  float* cfl  = ws + o; o += (size_t)BATCH * 2048;
  float* h1g  = ws + o; o += (size_t)BATCH * 256;
  float* corr = ws + o; o += (size_t)BATCH * CH;

  k_twiddles<<<32, 256, 0, stream>>>(Wy, WxA, ExA, Ey);
  k_fc0<<<(BATCH * NPIX + 255) / 256, 256, 0, stream>>>(xin, fc0w, fc0b, hA);

  float* hin = hA; float* hout = hB;
  for (int l = 0; l < 4; ++l) {
    // CFT correction branch
    k_cft<<<BATCH * CH, 256, 0, stream>>>(hin, cfl);
    k_g1<<<16, 32, 0, stream>>>(cfl, g1w + (size_t)l * 256 * 2048,
                                g1b + (size_t)l * 256, h1g);
    k_g2<<<1, 128, 0, stream>>>(h1g, g2w + (size_t)l * CH * 256,
                                g2b + (size_t)l * CH, corr);
    // spectral branch (partial DFTs as WMMA GEMMs)
    k_dft_y<<<1024, 128, 0, stream>>>(hin, Wy, yfb);
    k_dft_x<<<BATCH * CH, 128, 0, stream>>>(yfb, WxA, xfb);
    k_modemix<<<KX * KY, 32, 0, stream>>>(xfb,
        sw1r + (size_t)l * CH * CH * 256, sw1i + (size_t)l * CH * CH * 256,
        sw2r + (size_t)l * CH * CH * 256, sw2i + (size_t)l * CH * CH * 256, mfb);
    k_idft_x<<<BATCH * CH, 256, 0, stream>>>(mfb, ExA, zfb);
    // pointwise channel mixing
    k_chanmix<<<1024, 256, 0, stream>>>(hin, cw + (size_t)l * CH * CH,
                                        cb + (size_t)l * CH, h2b);
    // fused inverse-y DFT + residual + corr + gelu
    k_combine<<<1024, 128, 0, stream>>>(zfb, Ey, h2b, corr, hout, (l < 3) ? 1 : 0);
    float* t = hin; hin = hout; hout = t;
  }

  int mtiles = (NOUT + 15) / 16;
  k_head<<<(mtiles + 3) / 4, 128, 0, stream>>>(hin, fc1w, fc1b, fc2w, fc2b,
                                               (float*)d_out);
}